// ConcatAttentionUnit_57552561766571
// MI455X (gfx1250) — compile-verified
//
#include <hip/hip_runtime.h>
#include <hip/hip_bf16.h>
#include <stdint.h>

// ---------------------------------------------------------------------------
// ConcatAttentionUnit (Bahdanau additive attention) for MI455X / gfx1250.
//
//   sq = hq@W1, sp = hp@W2                          (WMMA f32 16x16x4)
//   s[b,p,q] = sum_v v[v]*tanh(sp[b,p,v]+sq[b,q,v]) (268M v_tanh_f32 — the wall)
//   a = softmax_q(s)
//   out[b] = a[b] @ hq[b]                           (WMMA f32 16x16x4)
//
// Roofline: ~13MB inputs + 8MB intermediates -> fully L2-resident (192MB);
// compute-bound on the transcendental loop. GEMMs (~2.1 GFLOP) are noise even
// on the f32 WMMA path, so we keep full f32 numerics. The hot loop stages sq
// tiles into LDS with CDNA5 async-to-LDS copies (ASYNCcnt), double-buffered.
// ---------------------------------------------------------------------------

#define B_DIM  4
#define Q_DIM  512
#define P_DIM  512
#define D_DIM  512   // Q_SIZE == P_SIZE
#define V_DIM  256
#define ROWS_PER_PASS 8                    // 8 wave32 per block, 1 q-row each
#define N_PASS (Q_DIM / ROWS_PER_PASS)     // 64

typedef __attribute__((ext_vector_type(2))) float v2f;
typedef __attribute__((ext_vector_type(8))) float v8f;

#if __has_builtin(__builtin_amdgcn_tanhf)
  #define FAST_TANH(x) __builtin_amdgcn_tanhf(x)
#elif __has_builtin(__builtin_amdgcn_tanh_f32)
  #define FAST_TANH(x) __builtin_amdgcn_tanh_f32(x)
#else
  #define FAST_TANH(x) tanhf(x)
#endif

// ---------------------------------------------------------------------------
// Generic batched f32 GEMM: C[b] = A[b] (MxK, lda) @ Bm[b] (KxN, ldb).
// One wave per 16x16 C tile, V_WMMA_F32_16X16X4_F32, K stepped by 4.
// VGPR layouts (ISA 7.12.2, 32-bit, wave32):
//   A 16x4 : lanes 0-15 -> K{0,1}, lanes 16-31 -> K{2,3}; M = lane&15
//   B 4x16 : VGPR0 rows K=0/K=2 per half, VGPR1 rows K=1/K=3; N = lane&15
//   C 16x16: VGPR j = row (j + 8*half), col = lane&15
// ---------------------------------------------------------------------------
__global__ __launch_bounds__(32)
void wmma_gemm_f32(const float* __restrict__ A,
                   const float* __restrict__ Bm,
                   float* __restrict__ C,
                   int K,
                   long batchStrideA, long batchStrideB, long batchStrideC,
                   int lda, int ldb, int ldc)
{
    const int  batch = blockIdx.z;
    A  += (long)batch * batchStrideA;
    Bm += (long)batch * batchStrideB;
    C  += (long)batch * batchStrideC;

    const int tileN = blockIdx.x * 16;
    const int tileM = blockIdx.y * 16;

    const int lane = threadIdx.x & 31;
    const int half = lane >> 4;     // K pair {0,1} vs {2,3}
    const int l16  = lane & 15;

    const float* __restrict__ Arow = A  + (long)(tileM + l16) * lda; // M = l16
    const float* __restrict__ Bcol = Bm + (tileN + l16);             // N = l16

    v8f acc = {};
#pragma unroll 4
    for (int k0 = 0; k0 < K; k0 += 4) {
        const int ka = k0 + 2 * half;
        v2f a, b;
        a.x = Arow[ka];
        a.y = Arow[ka + 1];
        b.x = Bcol[(long)ka * ldb];
        b.y = Bcol[(long)(ka + 1) * ldb];
        // 8 args: (neg_a, A, neg_b, B, c_mod, C, reuse_a, reuse_b)
        acc = __builtin_amdgcn_wmma_f32_16x16x4_f32(
                  false, a, false, b, (short)0, acc, false, false);
    }

    float* __restrict__ Cp = C + (long)(tileM + 8 * half) * ldc + tileN + l16;
#pragma unroll
    for (int j = 0; j < 8; ++j)
        Cp[(long)j * ldc] = acc[j];
}

// ---------------------------------------------------------------------------
// CDNA5 async global->LDS copy (ASYNCcnt tracked). Inline asm: portable
// across ROCm-7.2 / amdgpu-toolchain (the clang builtin arity differs).
// Each lane copies 16B to the LDS address in lds_dst and 16B to that address + 512 (IOFFSET applies
// to both global and LDS sides per CDNA5 ISA).
// ---------------------------------------------------------------------------
__device__ __forceinline__ void async_copy_row_chunk(uint32_t lds_dst,
                                                     const float* gsrc) {
    asm volatile("global_load_async_to_lds_b128 %0, %1, off"
                 :: "v"(lds_dst), "v"(gsrc) : "memory");
    asm volatile("global_load_async_to_lds_b128 %0, %1, off offset:512"
                 :: "v"(lds_dst), "v"(gsrc) : "memory");
}

// ---------------------------------------------------------------------------
// Scores + softmax. One 256-thread block (8 wave32) per (b,p).
// Per pass: 8 q-rows of sq (8KB) staged via async-to-LDS, double buffered;
// wave w reduces row q=pass*8+w (lane-strided over v, conflict-free), then
// block softmax over the 512 scores.
// ---------------------------------------------------------------------------
__global__ __launch_bounds__(256)
void scores_softmax_kernel(const float* __restrict__ sq,   // [B,Q,V]
                           const float* __restrict__ sp,   // [B,P,V]
                           const float* __restrict__ vvec, // [V]
                           float* __restrict__ attn)       // [B,P,Q]
{
    const int p    = blockIdx.x;
    const int b    = blockIdx.y;
    const int t    = threadIdx.x;
    const int w    = t >> 5;     // wave id 0..7
    const int lane = t & 31;

    __shared__ float sp_sm[V_DIM];
    __shared__ float vv_sm[V_DIM];
    __shared__ float sqtile[2][ROWS_PER_PASS * V_DIM];  // 2 x 8KB
    __shared__ float s_sm[Q_DIM];
    __shared__ float red[8];

    sp_sm[t] = sp[((long)b * P_DIM + p) * V_DIM + t];   // 256 threads == V_DIM
    vv_sm[t] = vvec[t];

    const float* __restrict__ sqb = sq + (long)b * Q_DIM * V_DIM;

    // Each thread moves 2x16B of row q = pass*8 + w: chunks lane and lane+32.
    auto issue = [&](int buf, int pass) {
        const int q = pass * ROWS_PER_PASS + w;
        const float* gsrc = sqb + (long)q * V_DIM + lane * 4;
        uint32_t dst = (uint32_t)(uintptr_t)(&sqtile[buf][w * V_DIM])
                       + (uint32_t)(lane * 16);
        async_copy_row_chunk(dst, gsrc);
    };

    issue(0, 0);

    for (int pass = 0; pass < N_PASS; ++pass) {
        const int cur = pass & 1;
        if (pass + 1 < N_PASS) {
            issue(cur ^ 1, pass + 1);
            // 4 ops outstanding; <=2 left  => current buffer's copies done
            asm volatile("s_wait_asynccnt 0x2" ::: "memory");
        } else {
            asm volatile("s_wait_asynccnt 0x0" ::: "memory");
        }
        __syncthreads();   // all waves' copies for `cur` visible (+ sp/vv on pass 0)

        const float* __restrict__ row = &sqtile[cur][w * V_DIM];
        float acc = 0.f;
#pragma unroll
        for (int j = 0; j < 8; ++j) {
            const int vi = lane + 32 * j;
            acc += vv_sm[vi] * FAST_TANH(row[vi] + sp_sm[vi]);
        }
        // wave32 sum
#pragma unroll
        for (int off = 16; off > 0; off >>= 1)
            acc += __shfl_xor(acc, off, 32);
        if (lane == 0) s_sm[pass * ROWS_PER_PASS + w] = acc;
        __syncthreads();   // row done before buffer `cur` is refilled next pass
    }

    // ---- softmax over the 512 scores (each thread owns q = t, t+256) ----
    const float s0 = s_sm[t];
    const float s1 = s_sm[t + 256];

    float wred = fmaxf(s0, s1);
#pragma unroll
    for (int off = 16; off > 0; off >>= 1)
        wred = fmaxf(wred, __shfl_xor(wred, off, 32));
    if (lane == 0) red[w] = wred;
    __syncthreads();
    float bmax = red[0];
#pragma unroll
    for (int i = 1; i < 8; ++i) bmax = fmaxf(bmax, red[i]);
    __syncthreads();

    const float e0 = __expf(s0 - bmax);
    const float e1 = __expf(s1 - bmax);
    float wsum = e0 + e1;
#pragma unroll
    for (int off = 16; off > 0; off >>= 1)
        wsum += __shfl_xor(wsum, off, 32);
    if (lane == 0) red[w] = wsum;
    __syncthreads();
    float bsum = red[0];
#pragma unroll
    for (int i = 1; i < 8; ++i) bsum += red[i];

    const float inv = 1.0f / bsum;
    float* __restrict__ arow = attn + ((long)b * P_DIM + p) * Q_DIM;
    arow[t]       = e0 * inv;
    arow[t + 256] = e1 * inv;
}

// ---------------------------------------------------------------------------
extern "C" void kernel_launch(void* const* d_in, const int* in_sizes, int n_in,
                              void* d_out, int out_size, void* d_ws, size_t ws_size,
                              hipStream_t stream) {
    const float* hq   = (const float*)d_in[0];  // [B,Q,D]
    const float* hp   = (const float*)d_in[1];  // [B,P,D]
    const float* W1   = (const float*)d_in[2];  // [D,V]
    const float* W2   = (const float*)d_in[3];  // [D,V]
    const float* vvec = (const float*)d_in[4];  // [V]
    float* out = (float*)d_out;                 // [B,P,D]

    // Workspace layout (8 MB total)
    float* sq   = (float*)d_ws;                       // B*Q*V
    float* sp   = sq + (long)B_DIM * Q_DIM * V_DIM;   // B*P*V
    float* attn = sp + (long)B_DIM * P_DIM * V_DIM;   // B*P*Q

    // 1) Projections, batch flattened into M (weights shared across batch).
    wmma_gemm_f32<<<dim3(V_DIM / 16, (B_DIM * Q_DIM) / 16, 1), 32, 0, stream>>>(
        hq, W1, sq, D_DIM, 0, 0, 0, D_DIM, V_DIM, V_DIM);
    wmma_gemm_f32<<<dim3(V_DIM / 16, (B_DIM * P_DIM) / 16, 1), 32, 0, stream>>>(
        hp, W2, sp, D_DIM, 0, 0, 0, D_DIM, V_DIM, V_DIM);

    // 2) tanh scores + softmax (async-to-LDS double-buffered hot loop)
    scores_softmax_kernel<<<dim3(P_DIM, B_DIM), 256, 0, stream>>>(sq, sp, vvec, attn);

    // 3) out[b] = attn[b] (512x512) @ hq[b] (512x512), batched over b
    wmma_gemm_f32<<<dim3(D_DIM / 16, P_DIM / 16, B_DIM), 32, 0, stream>>>(
        attn, hq, out, Q_DIM,
        (long)P_DIM * Q_DIM, (long)Q_DIM * D_DIM, (long)P_DIM * D_DIM,
        Q_DIM, D_DIM, D_DIM);
}